// Hyp_Siren_15745350108029
// MI455X (gfx1250) — compile-verified
//
#include <hip/hip_runtime.h>
#include <math.h>

typedef __attribute__((ext_vector_type(2))) float v2f;
typedef __attribute__((ext_vector_type(4))) float v4f;
typedef __attribute__((ext_vector_type(8))) float v8f;

#define BATCH   32
#define NPTS    16384
#define HID     256
#define ASTRIDE 260          /* padded LDS row stride (floats): conflict-free */
#define W_TOT   197888       /* 2*256 + 3*256*256 + 256*3 */
#define B_TOT   1027         /* 4*256 + 3 */
#define W_L1OFF 512          /* after layer-0 weights */
#define W_L4OFF 197120       /* 512 + 3*65536 */
#define OMEGA   30.0f

// Fast sin: Cody-Waite pi-reduction + odd Taylor poly (x..x^11).
// Abs error ~3e-7 for |x| < ~1e3; ~13 VALU ops, co-executes with WMMA.
__device__ __forceinline__ float fast_sin(float x) {
    const float INV_PI = 0.318309886183790672f;
    const float PI_HI  = 3.14159274101257324f;     // fp32(pi)
    const float PI_LO  = -8.74227765734758577e-8f; // pi - PI_HI
    float n = rintf(x * INV_PI);
    float f = fmaf(n, -PI_HI, x);
    f = fmaf(n, -PI_LO, f);                        // f in [-pi/2, pi/2]
    float f2 = f * f;
    float p = fmaf(f2, -2.50507034e-8f, 2.75573149e-6f);  // x^11, x^9
    p = fmaf(f2, p, -1.98412698e-4f);                     // x^7
    p = fmaf(f2, p,  8.33333333e-3f);                     // x^5
    p = fmaf(f2, p, -1.66666667e-1f);                     // x^3
    p = fmaf(f2 * f, p, f);
    return (((int)n) & 1) ? -p : p;                // sin(x) = (-1)^n sin(f)
}

// One wave (32 lanes) computes the full MLP for TWO 16-point tiles (32 pts).
// Every B-fragment load is shared by both tiles' WMMAs. Tile0's A lives in
// registers (128 VGPRs, in-place LDS update); tile1's A streams from LDS
// (ping-pong buffers). Four independent accumulation chains (tile x even/odd K)
// keep dependent WMMAs 4 instructions apart.
__global__ __launch_bounds__(32) void siren_wmma_kernel(
    const float* __restrict__ coords,   // [B, N, 2]
    const float* __restrict__ w,        // [B, W_TOT]
    const float* __restrict__ bia,      // [B, B_TOT]
    float* __restrict__ out)            // [B, N, 3]
{
    __shared__ float act0 [16 * ASTRIDE];   // tile0 (in-place)
    __shared__ float act1a[16 * ASTRIDE];   // tile1 ping
    __shared__ float act1b[16 * ASTRIDE];   // tile1 pong

    const int lane  = threadIdx.x & 31;
    const int blk   = blockIdx.x;              // 0 .. 16383
    const int batch = blk >> 9;                // 512 blocks per batch
    const int m0    = (blk & 511) << 5;        // first of 32 points

    const float* cB = coords + (size_t)batch * NPTS * 2;
    const float* wB = w      + (size_t)batch * W_TOT;
    const float* bB = bia    + (size_t)batch * B_TOT;

    const int hm = lane & 15;   // M (A) / N (B,D) index within 16
    const int hh = lane >> 4;   // K half-select: lanes 16-31 hold K+2,K+3

    // ---------------- layer 0: 2 -> 256 (tiny, VALU) ----------------
    for (int i = 0; i < 256; ++i) {
        int idx = lane + i * 32;         // 32*256 entries / 32 lanes
        int p = idx >> 8;                // 0..31 (uniform across lanes)
        int o = idx & 255;
        float c0 = cB[(m0 + p) * 2 + 0];
        float c1 = cB[(m0 + p) * 2 + 1];
        v2f  w0 = *(const v2f*)(wB + o * 2);
        float z  = fmaf(c0, w0.x, fmaf(c1, w0.y, bB[o]));
        float s  = fast_sin(OMEGA * z);
        if (p < 16) act0 [p * ASTRIDE + o]        = s;
        else        act1a[(p - 16) * ASTRIDE + o] = s;
    }
    // Same-wave LDS ops are in-order: later ds_loads see these stores.

    float* src1 = act1a;   // tile1 read buffer for this layer
    float* dst1 = act1b;   // tile1 write buffer for this layer

    // ---------------- hidden layers: 3 x (256 -> 256) via WMMA ----------------
    for (int L = 0; L < 3; ++L) {
        const float* wL = wB + W_L1OFF + L * (HID * HID);  // W[256][256] row-major
        const float* bL = bB + 256 + L * 256;

        // Phase A: cache tile0's A-fragments in registers (128 VGPRs).
        // A 16x4 layout: lane = M + 16*(K>=2); vgpr = K%2.
        v2f a0[64];
        const float* ar0 = &act0[hm * ASTRIDE + 2 * hh];
        #pragma unroll
        for (int kb = 0; kb < 64; ++kb) a0[kb] = *(const v2f*)(ar0 + 4 * kb);

        const float* ar1 = &src1[hm * ASTRIDE + 2 * hh];

        // Phase B: 16 output blocks of 16; accumulate K=256 in steps of 4.
        for (int nb = 0; nb < 16; ++nb) {
            const int n0 = nb * 16;
            v8f d0e = {0.f,0.f,0.f,0.f,0.f,0.f,0.f,0.f};
            v8f d0o = {0.f,0.f,0.f,0.f,0.f,0.f,0.f,0.f};
            v8f d1e = {0.f,0.f,0.f,0.f,0.f,0.f,0.f,0.f};
            v8f d1o = {0.f,0.f,0.f,0.f,0.f,0.f,0.f,0.f};
            // B[k][n] = W[n][k] -> contiguous float2 per lane per K-step,
            // shared by both tiles.
            const float* wN = wL + (n0 + hm) * HID + 2 * hh;
            #pragma unroll
            for (int kb = 0; kb < 64; kb += 2) {
                v2f bf0 = *(const v2f*)(wN + 4 * kb);        // global (L2/WGP$)
                v2f bf1 = *(const v2f*)(wN + 4 * kb + 4);
                v2f a10 = *(const v2f*)(ar1 + 4 * kb);       // LDS stream
                v2f a11 = *(const v2f*)(ar1 + 4 * kb + 4);
                d0e = __builtin_amdgcn_wmma_f32_16x16x4_f32(
                          false, a0[kb],   false, bf0, (short)0, d0e, false, false);
                d1e = __builtin_amdgcn_wmma_f32_16x16x4_f32(
                          false, a10,      false, bf0, (short)0, d1e, false, false);
                d0o = __builtin_amdgcn_wmma_f32_16x16x4_f32(
                          false, a0[kb+1], false, bf1, (short)0, d0o, false, false);
                d1o = __builtin_amdgcn_wmma_f32_16x16x4_f32(
                          false, a11,      false, bf1, (short)0, d1o, false, false);
            }
            // Bias + sin. Tile0 writes in place (A0 in regs); tile1 writes
            // the pong buffer (src still needed by later nb blocks).
            // C/D layout: vgpr v -> M = v + 8*hh, N = n0 + hm.
            float bv = bL[n0 + hm];
            #pragma unroll
            for (int v = 0; v < 8; ++v) {
                act0[(v + 8 * hh) * ASTRIDE + n0 + hm] =
                    fast_sin(OMEGA * ((d0e[v] + d0o[v]) + bv));
                dst1[(v + 8 * hh) * ASTRIDE + n0 + hm] =
                    fast_sin(OMEGA * ((d1e[v] + d1o[v]) + bv));
            }
        }
        { float* t = src1; src1 = dst1; dst1 = t; }   // swap ping-pong
        asm volatile("" ::: "memory");  // keep LDS store->load order across layers
    }

    // ---------------- final layer: 256 -> 3 (tiny, VALU, float4) ----------------
    const float* w4 = wB + W_L4OFF;     // W[3][256] row-major
    #pragma unroll
    for (int t = 0; t < 3; ++t) {
        int idx = lane + t * 32;        // 32 points * 3 outputs = 96 dots
        int p = idx / 3;                // 0..31
        int j = idx - p * 3;
        const float* wr = w4 + j * HID;                       // 16B aligned
        const float* ar = (p < 16 ? &act0[p * ASTRIDE]
                                  : &src1[(p - 16) * ASTRIDE]); // 16B aligned
        float s = bB[1024 + j];
        #pragma unroll 4
        for (int i = 0; i < HID; i += 4) {
            v4f av = *(const v4f*)(ar + i);
            v4f wv = *(const v4f*)(wr + i);
            s = fmaf(av.x, wv.x, s);
            s = fmaf(av.y, wv.y, s);
            s = fmaf(av.z, wv.z, s);
            s = fmaf(av.w, wv.w, s);
        }
        out[((size_t)batch * NPTS + m0 + p) * 3 + j] = s;
    }
}

extern "C" void kernel_launch(void* const* d_in, const int* in_sizes, int n_in,
                              void* d_out, int out_size, void* d_ws, size_t ws_size,
                              hipStream_t stream) {
    const float* coords = (const float*)d_in[0];   // [B, N, 2]
    const float* w      = (const float*)d_in[1];   // [B, W_TOT]
    const float* b      = (const float*)d_in[2];   // [B, B_TOT]
    const float* latent = (const float*)d_in[3];   // [B, 256]
    float* out = (float*)d_out;

    // d_out layout = (outputs, coords_HR, w, b, latent_space) concatenated flat.
    const size_t n_pts    = (size_t)in_sizes[0] / 2;   // B*N
    const size_t outElems = n_pts * 3;
    float* o_coords = out + outElems;
    float* o_w      = o_coords + in_sizes[0];
    float* o_b      = o_w + in_sizes[1];
    float* o_lat    = o_b + in_sizes[2];

    hipMemcpyAsync(o_coords, coords, (size_t)in_sizes[0] * sizeof(float),
                   hipMemcpyDeviceToDevice, stream);
    hipMemcpyAsync(o_w, w, (size_t)in_sizes[1] * sizeof(float),
                   hipMemcpyDeviceToDevice, stream);
    hipMemcpyAsync(o_b, b, (size_t)in_sizes[2] * sizeof(float),
                   hipMemcpyDeviceToDevice, stream);
    hipMemcpyAsync(o_lat, latent, (size_t)in_sizes[3] * sizeof(float),
                   hipMemcpyDeviceToDevice, stream);

    // 16384 blocks of one wave; each wave computes 32 points end-to-end.
    const int nBlocks = (BATCH * NPTS) / 32;   // 16384
    siren_wmma_kernel<<<nBlocks, 32, 0, stream>>>(coords, w, b, out);
}